// v_PT_13666585935889
// MI455X (gfx1250) — compile-verified
//
#include <hip/hip_runtime.h>
#include <stdint.h>

// CDNA5 async global->LDS staging (double-buffered) of gathered feature vectors.
#ifndef USE_ASYNC_STAGE
#define USE_ASYNC_STAGE 1
#endif

typedef float v4f __attribute__((ext_vector_type(4)));

namespace {
constexpr int Tn = 16, Xn = 16, Yn = 32, Zn = 32;
constexpr int NSITE  = Tn * Xn * Yn * Zn;   // 262144
constexpr int BLOCK  = 256;                 // 8 wave32 waves per workgroup
constexpr int GRIDX  = 256;
constexpr int STRIDE = GRIDX * BLOCK;       // 65536
constexpr int NITER  = NSITE / STRIDE;      // 4 (exact)
static_assert(NSITE % STRIDE == 0, "trip count must be exact");
}

// Periodic-BC neighbor on the 16x16x32x32 lattice (pow2 dims -> mask wrap).
// mu is compile-time constant in every instantiation -> folds to shift/mask.
__device__ __forceinline__ int neighbor(int s, int mu, int d) {
  int t = (s >> 14) & 15;
  int x = (s >> 10) & 15;
  int y = (s >> 5) & 31;
  int z = s & 31;
  if      (mu == 0) t = (t + d) & 15;
  else if (mu == 1) x = (x + d) & 15;
  else if (mu == 2) y = (y + d) & 31;
  else              z = (z + d) & 31;
  return (t << 14) | (x << 10) | (y << 5) | z;
}

__device__ __forceinline__ void mv(const float u[9], const float* vi, float* vo) {
  vo[0] = fmaf(u[0], vi[0], fmaf(u[1], vi[1], u[2] * vi[2]));
  vo[1] = fmaf(u[3], vi[0], fmaf(u[4], vi[1], u[5] * vi[2]));
  vo[2] = fmaf(u[6], vi[0], fmaf(u[7], vi[1], u[8] * vi[2]));
}
__device__ __forceinline__ void mvT(const float u[9], const float* vi, float* vo) {
  vo[0] = fmaf(u[0], vi[0], fmaf(u[3], vi[1], u[6] * vi[2]));
  vo[1] = fmaf(u[1], vi[0], fmaf(u[4], vi[1], u[7] * vi[2]));
  vo[2] = fmaf(u[2], vi[0], fmaf(u[5], vi[1], u[8] * vi[2]));
}

// Per-path source-site / link-matrix addressing (compile-time geometry).
template <int P>
__device__ __forceinline__ const float* path_addrs(
    const float* __restrict__ finp, const float* __restrict__ Ug, int site,
    const float*& Ua, const float*& Ub) {
  int src;
  if constexpr (P == 9) {  // [(0,+1),(1,+1)]: U1(x) U0(x+x^) v(x+x^+t^)
    const int sx = neighbor(site, 1, +1);
    src = neighbor(sx, 0, +1);
    Ua  = Ug + ((size_t)0 * NSITE + sx) * 9;
    Ub  = Ug + ((size_t)1 * NSITE + site) * 9;
  } else {
    constexpr int  MU  = (P - 1) / 2;
    constexpr bool FWD = (P & 1) != 0;
    if constexpr (FWD) {   // v'(x) = U_mu(x) v(x+mu)
      src = neighbor(site, MU, +1);
      Ua  = Ug + ((size_t)MU * NSITE + site) * 9;
    } else {               // v'(x) = U_mu(x-mu)^T v(x-mu)
      src = neighbor(site, MU, -1);
      Ua  = Ug + ((size_t)MU * NSITE + src) * 9;
    }
    Ub = nullptr;
  }
  return finp + (size_t)src * 12;  // 48 B per site, 16B aligned
}

#if USE_ASYNC_STAGE
// CDNA5 async DMA: 48 B/lane global -> this lane's LDS slot, non-temporal
// (features are read exactly once; keep the 192MB L2 for the reused U links).
__device__ __forceinline__ void async_gather48(uint32_t lds_slot,
                                               const float* vsrc) {
  asm volatile(
      "global_load_async_to_lds_b128 %0, %1, off th:TH_LOAD_NT\n\t"
      "global_load_async_to_lds_b128 %0, %1, off offset:16 th:TH_LOAD_NT\n\t"
      "global_load_async_to_lds_b128 %0, %1, off offset:32 th:TH_LOAD_NT"
      :: "v"(lds_slot), "v"(vsrc) : "memory");
}
#endif

// One kernel instantiation per path: all control flow is compile-time.
template <int P>
__global__ __launch_bounds__(BLOCK) void v_pt_kernel(
    const float* __restrict__ fin,   // [10, NSITE, 4, 3]
    const float* __restrict__ Ug,    // [4,  NSITE, 3, 3]
    float* __restrict__ out)         // [10, NSITE, 4, 3]
{
  const float* __restrict__ finp = fin + (size_t)P * NSITE * 12;
  float* __restrict__ outp       = out + (size_t)P * NSITE * 12;
  const int tid0 = (int)(blockIdx.x * BLOCK + threadIdx.x);

  if constexpr (P == 0) {
    // Identity path: pure NT b128 stream copy, exact unrolled trip count.
#pragma unroll
    for (int k = 0; k < NITER; ++k) {
      const int site = tid0 + k * STRIDE;
      const v4f* v4 = (const v4f*)(finp + (size_t)site * 12);
      v4f*       o4 = (v4f*)(outp + (size_t)site * 12);
      v4f a = __builtin_nontemporal_load(v4 + 0);
      v4f b = __builtin_nontemporal_load(v4 + 1);
      v4f c = __builtin_nontemporal_load(v4 + 2);
      __builtin_nontemporal_store(a, o4 + 0);
      __builtin_nontemporal_store(b, o4 + 1);
      __builtin_nontemporal_store(c, o4 + 2);
    }
  } else {
#if USE_ASYNC_STAGE
    // Double-buffered LDS staging: DMA for site k+1 overlaps compute of site k.
    __shared__ __align__(16) float stage[2][BLOCK * 12];  // 2 x 48 B per lane
    const uint32_t slot0 = (uint32_t)(uintptr_t)&stage[0][threadIdx.x * 12];
    const uint32_t slot1 = (uint32_t)(uintptr_t)&stage[1][threadIdx.x * 12];

    const float *Ua, *Ub;
    const float* vsrc = path_addrs<P>(finp, Ug, tid0, Ua, Ub);
    async_gather48(slot0, vsrc);  // prologue: stage iteration 0

#pragma unroll
    for (int k = 0; k < NITER; ++k) {
      const int site = tid0 + k * STRIDE;
      const float* UaC = Ua;
      const float* UbC = Ub;

      // Stage iteration k+1 into the other buffer before waiting on k.
      if (k + 1 < NITER) {
        vsrc = path_addrs<P>(finp, Ug, site + STRIDE, Ua, Ub);
        async_gather48((k & 1) ? slot0 : slot1, vsrc);
        __builtin_prefetch(Ua, 0, 0);  // warm next links (global_prefetch_b8)
      }

      // Current links from L2 while both DMAs are in flight.
      float ua[9], ub[9];
#pragma unroll
      for (int j = 0; j < 9; ++j) ua[j] = UaC[j];
      if constexpr (P == 9) {
#pragma unroll
        for (int j = 0; j < 9; ++j) ub[j] = UbC[j];
      }

      // Async loads complete in order: <=3 outstanding => iteration k done.
      if (k + 1 < NITER) asm volatile("s_wait_asynccnt 0x3" ::: "memory");
      else               asm volatile("s_wait_asynccnt 0x0" ::: "memory");

      float vin[12];
      {
        const v4f* s4 = (const v4f*)&stage[k & 1][threadIdx.x * 12];
        v4f a = s4[0], b = s4[1], c = s4[2];
        vin[0]=a[0]; vin[1]=a[1]; vin[2]=a[2];  vin[3]=a[3];
        vin[4]=b[0]; vin[5]=b[1]; vin[6]=b[2];  vin[7]=b[3];
        vin[8]=c[0]; vin[9]=c[1]; vin[10]=c[2]; vin[11]=c[3];
      }
#else
    const float *Ua, *Ub;
#pragma unroll
    for (int k = 0; k < NITER; ++k) {
      const int site = tid0 + k * STRIDE;
      const float* vsrc = path_addrs<P>(finp, Ug, site, Ua, Ub);
      const float* UaC = Ua;
      const float* UbC = Ub;
      float vin[12];
      {
        const v4f* v4 = (const v4f*)vsrc;
        v4f a = __builtin_nontemporal_load(v4 + 0);
        v4f b = __builtin_nontemporal_load(v4 + 1);
        v4f c = __builtin_nontemporal_load(v4 + 2);
        vin[0]=a[0]; vin[1]=a[1]; vin[2]=a[2];  vin[3]=a[3];
        vin[4]=b[0]; vin[5]=b[1]; vin[6]=b[2];  vin[7]=b[3];
        vin[8]=c[0]; vin[9]=c[1]; vin[10]=c[2]; vin[11]=c[3];
      }
      float ua[9], ub[9];
#pragma unroll
      for (int j = 0; j < 9; ++j) ua[j] = UaC[j];
      if constexpr (P == 9) {
#pragma unroll
        for (int j = 0; j < 9; ++j) ub[j] = UbC[j];
      }
#endif

      // ---- per-spin 3x3 color matvec(s), transpose chosen at compile time
      float res[12];
      if constexpr (P == 9) {
        float tmp[12];
#pragma unroll
        for (int s = 0; s < 4; ++s) mv(ua, vin + 3 * s, tmp + 3 * s);
#pragma unroll
        for (int s = 0; s < 4; ++s) mv(ub, tmp + 3 * s, res + 3 * s);
      } else if constexpr ((P & 1) != 0) {
#pragma unroll
        for (int s = 0; s < 4; ++s) mv(ua, vin + 3 * s, res + 3 * s);
      } else {
#pragma unroll
        for (int s = 0; s < 4; ++s) mvT(ua, vin + 3 * s, res + 3 * s);
      }

      // ---- non-temporal streaming store (write-once, never re-read) ----
      v4f r0 = {res[0], res[1], res[2],  res[3]};
      v4f r1 = {res[4], res[5], res[6],  res[7]};
      v4f r2 = {res[8], res[9], res[10], res[11]};
      v4f* o4 = (v4f*)(outp + (size_t)site * 12);
      __builtin_nontemporal_store(r0, o4 + 0);
      __builtin_nontemporal_store(r1, o4 + 1);
      __builtin_nontemporal_store(r2, o4 + 2);
    }
  }
}

extern "C" void kernel_launch(void* const* d_in, const int* in_sizes, int n_in,
                              void* d_out, int out_size, void* d_ws,
                              size_t ws_size, hipStream_t stream) {
  const float* fin = (const float*)d_in[0];  // [10,16,16,32,32,4,3] f32
  const float* U   = (const float*)d_in[1];  // [4,16,16,32,32,3,3]  f32
  float* out       = (float*)d_out;          // [10,16,16,32,32,4,3] f32

  dim3 grid(GRIDX, 1, 1);
  dim3 block(BLOCK, 1, 1);
#define LAUNCH_PATH(P) \
  v_pt_kernel<P><<<grid, block, 0, stream>>>(fin, U, out)
  LAUNCH_PATH(0); LAUNCH_PATH(1); LAUNCH_PATH(2); LAUNCH_PATH(3);
  LAUNCH_PATH(4); LAUNCH_PATH(5); LAUNCH_PATH(6); LAUNCH_PATH(7);
  LAUNCH_PATH(8); LAUNCH_PATH(9);
#undef LAUNCH_PATH

  (void)in_sizes; (void)n_in; (void)out_size; (void)d_ws; (void)ws_size;
}